// VanillaSSM_52682068852950
// MI455X (gfx1250) — compile-verified
//
#include <hip/hip_runtime.h>
#include <hip/hip_bf16.h>
#include <stdint.h>

typedef __attribute__((ext_vector_type(2))) float v2f;
typedef __attribute__((ext_vector_type(8))) float v8f;

constexpr int BSZ  = 32;
constexpr int LEN  = 8192;
constexpr int NST  = 256;
constexpr int SEG  = 128;
constexpr int NSEG = LEN / SEG; // 64
constexpr int XROW = SEG + 1;   // padded LDS row stride (TDM pad: +1 dword / 128 dwords)

// ---- workspace layout (float offsets) ----
constexpr size_t SZ_W   = (size_t)(SEG + 1) * NST;   // 129*256
constexpr size_t SZ_U   = (size_t)NST * SEG;         // 256*128
constexpr size_t SZ_KT  = (size_t)SEG * SEG;         // 128*128
constexpr size_t SZ_S   = (size_t)NSEG * NST * BSZ;  // 64*256*32
constexpr size_t OFF_WR = 0;
constexpr size_t OFF_WI = OFF_WR + SZ_W;
constexpr size_t OFF_UR = OFF_WI + SZ_W;
constexpr size_t OFF_UI = OFF_UR + SZ_U;
constexpr size_t OFF_AR = OFF_UI + SZ_U;
constexpr size_t OFF_AI = OFF_AR + NST;
constexpr size_t OFF_KT = OFF_AI + NST;
constexpr size_t OFF_SR = OFF_KT + SZ_KT;   // chunk-local state (re) -> overwritten with h_in (re)
constexpr size_t OFF_SI = OFF_SR + SZ_S;    // same for imag
// total ~1.2M floats (~4.8 MB) of d_ws

// ---------- TDM (Tensor Data Mover) helper ----------
#if __has_builtin(__builtin_amdgcn_tensor_load_to_lds)
#define HAVE_TDM 1
#else
#define HAVE_TDM 0
#endif

#if HAVE_TDM
typedef unsigned int v4u __attribute__((ext_vector_type(4)));
typedef int v4i __attribute__((ext_vector_type(4)));
typedef int v8i __attribute__((ext_vector_type(8)));

// 2D tile load: tile (tl0 x tl1) of 4-byte elements from row-major tensor with
// dim0 stride `stride0` (elements) into LDS at byte offset lds_off.
// padw0 = pad_enable/pad_interval/pad_amount bits of D# group1 word0.
__device__ __forceinline__ void tdm_load_2d(unsigned lds_off, const void* gptr,
                                            unsigned td0, unsigned td1,
                                            unsigned tl0, unsigned tl1,
                                            unsigned long long stride0,
                                            unsigned padw0) {
  unsigned long long ga = (unsigned long long)(uintptr_t)gptr;
  v4u g0;
  g0.x = 1u;                                              // count=1, is_restore=0, gather off
  g0.y = lds_off;                                         // lds_addr (bytes)
  g0.z = (unsigned)ga;                                    // global_addr[31:0]
  g0.w = (unsigned)((ga >> 32) & 0x1FFFFFFu) | (2u << 30); // global_addr[56:32] | type=2
  v8i g1;
  g1[0] = (int)((2u << 16) | padw0);                      // data_size=4B | pad fields
  g1[1] = (int)(td0 << 16);                               // tensor_dim0[15:0] @ bits63:48
  g1[2] = (int)((td0 >> 16) | (td1 << 16));               // tensor_dim0[31:16] | tensor_dim1[15:0]
  g1[3] = (int)((td1 >> 16) | (tl0 << 16));               // tensor_dim1[31:16] | tile_dim0
  g1[4] = (int)(tl1 & 0xFFFFu);                           // tile_dim1 (tile_dim2=0)
  g1[5] = (int)(unsigned)(stride0 & 0xFFFFFFFFu);         // tensor_dim0_stride[31:0]
  g1[6] = (int)(unsigned)((stride0 >> 32) & 0xFFFFu);     // tensor_dim0_stride[47:32]
  g1[7] = 0;
  v4i gz4 = {0, 0, 0, 0};
  v8i gz8 = {0, 0, 0, 0, 0, 0, 0, 0};
  // 6-arg form (clang-23 / therock-10.0 headers): groups 0..3 + extra group + cpol
  __builtin_amdgcn_tensor_load_to_lds(g0, g1, gz4, gz4, gz8, 0);
}
#endif

// ---------- WMMA helpers (V_WMMA_F32_16X16X4_F32, wave32) ----------
// A 16x4 f32: lane holds M=lane%16; VGPR v holds K = v + 2*(lane/16)
__device__ __forceinline__ v2f frag_a(const float* A, int lda, int m0, int k0) {
  int lane = threadIdx.x & 31;
  const float* p = A + (size_t)(m0 + (lane & 15)) * lda + k0 + ((lane >> 4) << 1);
  v2f a; a.x = p[0]; a.y = p[1]; return a;
}
// B 4x16 from [K][N] row-major (ldb = N-count): element (k,n) = Bm[k*ldb + n]
__device__ __forceinline__ v2f frag_b(const float* Bm, int ldb, int k0, int n0) {
  int lane = threadIdx.x & 31;
  int n = n0 + (lane & 15);
  int k = k0 + ((lane >> 4) << 1);
  v2f b; b.x = Bm[(size_t)k * ldb + n]; b.y = Bm[(size_t)(k + 1) * ldb + n]; return b;
}
// B 4x16 from [N][K] row-major (ldb = K row stride): element (k,n) = Bm[n*ldb + k]
__device__ __forceinline__ v2f frag_bT(const float* Bm, int ldb, int k0, int n0) {
  int lane = threadIdx.x & 31;
  int n = n0 + (lane & 15);
  int k = k0 + ((lane >> 4) << 1);
  const float* p = Bm + (size_t)n * ldb + k;
  v2f b; b.x = p[0]; b.y = p[1]; return b;
}
__device__ __forceinline__ v8f wmma4(v2f a, v2f b, v8f c) {
  return __builtin_amdgcn_wmma_f32_16x16x4_f32(false, a, false, b, (short)0, c,
                                               false, false);
}

// ---------- P1: per-(d,n) tables: W = C*A^d, U = A^(SEG-1-s)*B, A^SEG ----------
__global__ void p1_tables(const float* __restrict__ rad_log, const float* __restrict__ angle,
                          const float* __restrict__ Br, const float* __restrict__ Bi,
                          const float* __restrict__ Cr, const float* __restrict__ Ci,
                          float* __restrict__ ws) {
  int d = blockIdx.x;   // 0..SEG
  int n = threadIdx.x;  // 0..NST-1
  float e  = expf(rad_log[n]);
  float r  = expf(-(float)d * e);          // radius^d
  float th = (float)d * angle[n];
  float cs = cosf(th), sn = sinf(th);
  float apr = r * cs, api = r * sn;        // A^d
  float cr = Cr[n], ci = Ci[n];
  ws[OFF_WR + (size_t)d * NST + n] = cr * apr - ci * api;  // Re(C A^d)
  ws[OFF_WI + (size_t)d * NST + n] = cr * api + ci * apr;  // Im(C A^d)
  if (d < SEG) {
    int s = SEG - 1 - d;                   // exponent SEG-1-s == d
    float br = Br[n], bi = Bi[n];
    ws[OFF_UR + (size_t)n * SEG + s] = apr * br - api * bi;
    ws[OFF_UI + (size_t)n * SEG + s] = apr * bi + api * br;
  }
  if (d == SEG) { ws[OFF_AR + n] = apr; ws[OFF_AI + n] = api; }
}

// ---------- P2: k[tau] = sum_n Re(C A^tau B); materialize lower-tri Toeplitz ----------
__global__ void p2_toeplitz(const float* __restrict__ Br, const float* __restrict__ Bi,
                            float* __restrict__ ws) {
  int tau = blockIdx.x;  // 0..SEG-1
  int n   = threadIdx.x; // 0..255
  __shared__ float red[NST];
  float wr = ws[OFF_WR + (size_t)tau * NST + n];
  float wi = ws[OFF_WI + (size_t)tau * NST + n];
  red[n] = wr * Br[n] - wi * Bi[n];
  __syncthreads();
  for (int s = NST / 2; s > 0; s >>= 1) {
    if (n < s) red[n] += red[n + s];
    __syncthreads();
  }
  float kv = red[0];
  float* Kt = ws + OFF_KT;
  if (n < SEG) {
    int t = n + tau;
    if (t < SEG) {
      Kt[(size_t)t * SEG + n] = kv;                     // lower diagonal tau
      if (tau > 0) Kt[(size_t)n * SEG + t] = 0.f;       // upper diagonal tau -> 0
    }
  }
}

// ---------- P3: per-chunk  Y_local = Ktoe . x^T ; S = U . x^T  (WMMA) ----------
__global__ void p3_local(const float* __restrict__ x, float* __restrict__ out,
                         float* __restrict__ ws) {
  int g  = blockIdx.x;
  int t0 = g * SEG;
  const float* Kt = ws + OFF_KT;
  const float* Ur = ws + OFF_UR;
  const float* Ui = ws + OFF_UI;
  float* SR = ws + OFF_SR + (size_t)g * NST * BSZ;
  float* SI = ws + OFF_SI + (size_t)g * NST * BSZ;

  // x tile staged as [b][s] with padded row stride XROW=129 (bank-conflict-free B frags)
  __shared__ __align__(16) float xs[BSZ * XROW];
#if HAVE_TDM
  if (threadIdx.x < 32) {  // wave 0 issues the DMA; TDM ignores EXEC
    tdm_load_2d((unsigned)(uintptr_t)&xs[0], x + t0,
                /*tensor*/ LEN, BSZ, /*tile*/ SEG, BSZ, /*stride*/ LEN,
                /*pad: enable, interval=128 dwords, amount=1 dword*/
                (1u << 20) | (6u << 22) | (0u << 25));
    __builtin_amdgcn_s_wait_tensorcnt(0);
  }
#else
  for (int i = threadIdx.x; i < BSZ * SEG; i += blockDim.x) {
    int b = i >> 7;
    int s = i & (SEG - 1);
    xs[(size_t)b * XROW + s] = x[(size_t)b * LEN + t0 + s];
  }
#endif
  __syncthreads();

  int w    = threadIdx.x >> 5;
  int lane = threadIdx.x & 31;
  // 80 output tiles: 16 for Y_local (128x32), 32 for S_re (256x32), 32 for S_im
  for (int tile = w; tile < 80; tile += 8) {
    const float* A;
    int mt, nt, kind;
    if (tile < 16)      { kind = 0; mt = tile >> 1;        nt = tile & 1;        A = Kt; }
    else if (tile < 48) { kind = 1; int u = tile - 16; mt = u >> 1; nt = u & 1;  A = Ur; }
    else                { kind = 2; int u = tile - 48; mt = u >> 1; nt = u & 1;  A = Ui; }
    v8f c = {};
    for (int kk = 0; kk < SEG; kk += 4) {
      v2f a = frag_a(A, SEG, mt * 16, kk);
      v2f b = frag_bT(xs, XROW, kk, nt * 16);
      c = wmma4(a, b, c);
    }
    int nc  = nt * 16 + (lane & 15);          // column (batch)
    int mrb = mt * 16 + ((lane >> 4) << 3);   // row base
    if (kind == 0) {
      for (int v = 0; v < 8; ++v)
        out[(size_t)nc * LEN + t0 + mrb + v] = 2.0f * c[v];
    } else if (kind == 1) {
      for (int v = 0; v < 8; ++v) SR[(size_t)(mrb + v) * BSZ + nc] = c[v];
    } else {
      for (int v = 0; v < 8; ++v) SI[(size_t)(mrb + v) * BSZ + nc] = c[v];
    }
  }
}

// ---------- P4: 64-step inter-chunk diagonal scan; converts S[g] -> h_in[g] in place ----------
__global__ void p4_scan(float* __restrict__ ws) {
  int id = blockIdx.x * blockDim.x + threadIdx.x;  // 0..8191
  int b  = id & (BSZ - 1);
  int n  = id >> 5;
  float ar = ws[OFF_AR + n], ai = ws[OFF_AI + n];  // A^SEG
  float hr = 0.f, hi = 0.f;
  size_t base = (size_t)n * BSZ + b;
  for (int g = 0; g < NSEG; ++g) {
    size_t idx = (size_t)g * NST * BSZ + base;
    float sr = ws[OFF_SR + idx];
    float si = ws[OFF_SI + idx];
    ws[OFF_SR + idx] = hr;   // h entering chunk g
    ws[OFF_SI + idx] = hi;
    float nhr = ar * hr - ai * hi + sr;
    float nhi = ar * hi + ai * hr + si;
    hr = nhr; hi = nhi;
  }
}

// ---------- P5: per-chunk cross term  Y += Re(C A^{j+1} h_in)  (WMMA) ----------
__global__ void p5_cross(float* __restrict__ out, const float* __restrict__ ws) {
  int g  = blockIdx.x;
  int t0 = g * SEG;
  const float* Wr = ws + OFF_WR;
  const float* Wi = ws + OFF_WI;
  const float* HR = ws + OFF_SR + (size_t)g * NST * BSZ;
  const float* HI = ws + OFF_SI + (size_t)g * NST * BSZ;

  __shared__ __align__(16) float hs[2 * NST * BSZ];  // 64 KB: h_in re | im, [n][b]
#if HAVE_TDM
  if (threadIdx.x < 32) {  // wave 0 issues both DMAs (1D copies, 32 KB each)
    tdm_load_2d((unsigned)(uintptr_t)&hs[0], HR,
                NST * BSZ, 1, NST * BSZ, 1, NST * BSZ, 0u);
    tdm_load_2d((unsigned)(uintptr_t)&hs[0] + NST * BSZ * 4u, HI,
                NST * BSZ, 1, NST * BSZ, 1, NST * BSZ, 0u);
    __builtin_amdgcn_s_wait_tensorcnt(0);
  }
#else
  for (int i = threadIdx.x; i < NST * BSZ; i += blockDim.x) {
    hs[i] = HR[i];
    hs[NST * BSZ + i] = HI[i];
  }
#endif
  __syncthreads();

  int w    = threadIdx.x >> 5;
  int lane = threadIdx.x & 31;
  for (int tile = w; tile < 16; tile += 8) {  // 128x32 output, 2 tiles per wave
    int mt = tile >> 1, nt = tile & 1;
    v8f c = {};
    // + Wr[j+1,:] . h_re
    for (int kk = 0; kk < NST; kk += 4) {
      int m = mt * 16 + (lane & 15);
      int k = kk + ((lane >> 4) << 1);
      const float* p = Wr + (size_t)(m + 1) * NST + k;
      v2f a; a.x = p[0]; a.y = p[1];
      v2f b = frag_b(hs, BSZ, kk, nt * 16);
      c = wmma4(a, b, c);
    }
    // - Wi[j+1,:] . h_im
    for (int kk = 0; kk < NST; kk += 4) {
      int m = mt * 16 + (lane & 15);
      int k = kk + ((lane >> 4) << 1);
      const float* p = Wi + (size_t)(m + 1) * NST + k;
      v2f a; a.x = -p[0]; a.y = -p[1];
      v2f b = frag_b(hs + NST * BSZ, BSZ, kk, nt * 16);
      c = wmma4(a, b, c);
    }
    int nc  = nt * 16 + (lane & 15);
    int mrb = mt * 16 + ((lane >> 4) << 3);
    for (int v = 0; v < 8; ++v)
      out[(size_t)nc * LEN + t0 + mrb + v] += 2.0f * c[v];
  }
}

extern "C" void kernel_launch(void* const* d_in, const int* in_sizes, int n_in,
                              void* d_out, int out_size, void* d_ws, size_t ws_size,
                              hipStream_t stream) {
  (void)in_sizes; (void)n_in; (void)out_size; (void)ws_size;
  const float* x       = (const float*)d_in[0];
  const float* rad_log = (const float*)d_in[1];
  const float* angle   = (const float*)d_in[2];
  const float* Br      = (const float*)d_in[3];
  const float* Bi      = (const float*)d_in[4];
  const float* Cr      = (const float*)d_in[5];
  const float* Ci      = (const float*)d_in[6];
  float* out = (float*)d_out;
  float* ws  = (float*)d_ws;

  p1_tables<<<SEG + 1, NST, 0, stream>>>(rad_log, angle, Br, Bi, Cr, Ci, ws);
  p2_toeplitz<<<SEG, NST, 0, stream>>>(Br, Bi, ws);
  p3_local<<<NSEG, 256, 0, stream>>>(x, out, ws);
  p4_scan<<<BSZ, 256, 0, stream>>>(ws);
  p5_cross<<<NSEG, 256, 0, stream>>>(out, ws);
}